// MSAInitializer_66984309948701
// MI455X (gfx1250) — compile-verified
//
#include <hip/hip_runtime.h>

typedef float v2f __attribute__((ext_vector_type(2)));
typedef float v8f __attribute__((ext_vector_type(8)));
typedef float f4  __attribute__((ext_vector_type(4)));

#define LSEQ     1024
#define N_MSA    256
#define SEQ_DIM  64
#define MSA_IN   32
#define MSA_EMB  256
#define PAIR_DIM 128
#define NUM_BITS 14
#define POS_BINS 65

// ---------------------------------------------------------------------------
// Precompute: combined[l,c] = seq[l]@W_s + pos1d[l]@W_pos2 + b_s + b_pos2 + b_msa
// ---------------------------------------------------------------------------
__global__ __launch_bounds__(256) void combined_kernel(
    const float* __restrict__ seq, const float* __restrict__ Ws,
    const float* __restrict__ bs, const float* __restrict__ Wpos2,
    const float* __restrict__ bpos2, const float* __restrict__ bmsa,
    float* __restrict__ combined)
{
    const int l = blockIdx.x;
    const int c = threadIdx.x;
    float acc = bs[c] + bpos2[c] + bmsa[c];
#pragma unroll 8
    for (int s = 0; s < SEQ_DIM; ++s)
        acc = fmaf(seq[l * SEQ_DIM + s], Ws[s * MSA_EMB + c], acc);
#pragma unroll
    for (int b = 0; b < NUM_BITS; ++b)
        if ((l >> b) & 1) acc += Wpos2[b * MSA_EMB + c];
    combined[l * MSA_EMB + c] = acc;
}

// ---------------------------------------------------------------------------
// Precompute: q = seq@W_q + b_q ; k = seq@W_k + b_k
// ---------------------------------------------------------------------------
__global__ __launch_bounds__(128) void qk_kernel(
    const float* __restrict__ seq,
    const float* __restrict__ Wq, const float* __restrict__ bq,
    const float* __restrict__ Wk, const float* __restrict__ bk,
    float* __restrict__ q, float* __restrict__ k)
{
    const int l = blockIdx.x;
    const int c = threadIdx.x;
    float aq = bq[c], ak = bk[c];
#pragma unroll 8
    for (int s = 0; s < SEQ_DIM; ++s) {
        const float sv = seq[l * SEQ_DIM + s];
        aq = fmaf(sv, Wq[s * PAIR_DIM + c], aq);
        ak = fmaf(sv, Wk[s * PAIR_DIM + c], ak);
    }
    q[l * PAIR_DIM + c] = aq;
    k[l * PAIR_DIM + c] = ak;
}

// ---------------------------------------------------------------------------
// Precompute: posrow[r,c] = W_pos[r,c] + b_pos[c]
// ---------------------------------------------------------------------------
__global__ __launch_bounds__(128) void posrow_kernel(
    const float* __restrict__ Wpos, const float* __restrict__ bpos,
    float* __restrict__ posrow)
{
    const int r = blockIdx.x;
    const int c = threadIdx.x;
    posrow[r * PAIR_DIM + c] = Wpos[r * PAIR_DIM + c] + bpos[c];
}

// ---------------------------------------------------------------------------
// MSA main: out[g, :] = msa[g, 0:32] @ W_msa + combined[g % 1024, :]
// One wave = 16 flattened rows x 256 cols; K=32 via 8x V_WMMA_F32_16X16X4_F32.
// A layout: lanes 0-15 = M rows, VGPR pair = {K, K+1}, half-wave picks K+0/K+2.
// B/C/D layout: lane = N column (mod 16), VGPR index = K row / M row.
// ---------------------------------------------------------------------------
__global__ __launch_bounds__(256) void msa_wmma_kernel(
    const float* __restrict__ A, const float* __restrict__ W,
    const float* __restrict__ combined, float* __restrict__ out)
{
    __shared__ float Wlds[MSA_IN * MSA_EMB];  // 32 KB, whole W_msa
    {
        const f4* src = (const f4*)W;
        f4* dst = (f4*)Wlds;
#pragma unroll
        for (int i = 0; i < (MSA_IN * MSA_EMB / 4) / 256; ++i)
            dst[threadIdx.x + i * 256] = src[threadIdx.x + i * 256];
    }
    __syncthreads();

    const int tid    = threadIdx.x;
    const int wave   = tid >> 5;
    const int lane   = tid & 31;
    const int laneM  = lane & 15;
    const int laneHi = lane >> 4;           // 0: K+{0,1}, 1: K+{2,3}

    const int rowTile = blockIdx.x * 8 + wave;          // 2048 blocks * 8 waves = 16384 tiles
    const int g0 = rowTile * 16;                        // 16 flattened rows
    const int l0 = g0 & (LSEQ - 1);                     // tiles never cross the L boundary

    // A operand: 8 K-chunks of 4, each lane holds 2 consecutive floats
    v2f a[8];
    const float* Arow = A + (size_t)(g0 + laneM) * MSA_IN;
#pragma unroll
    for (int kc = 0; kc < 8; ++kc)
        a[kc] = *(const v2f*)(Arow + kc * 4 + laneHi * 2);

    for (int ct = 0; ct < MSA_EMB / 16; ++ct) {
        const int cb = ct * 16 + laneM;

        v2f b[8];
#pragma unroll
        for (int kc = 0; kc < 8; ++kc) {
            const int kb = kc * 4 + laneHi * 2;
            b[kc].x = Wlds[kb * MSA_EMB + cb];
            b[kc].y = Wlds[(kb + 1) * MSA_EMB + cb];
        }

        // Accumulator pre-loaded with combined[] -> broadcast add is free
        v8f c;
#pragma unroll
        for (int r = 0; r < 8; ++r)
            c[r] = combined[(size_t)(l0 + laneHi * 8 + r) * MSA_EMB + cb];

#pragma unroll
        for (int kc = 0; kc < 8; ++kc)
            c = __builtin_amdgcn_wmma_f32_16x16x4_f32(
                    false, a[kc], false, b[kc], (short)0, c, false, false);

#pragma unroll
        for (int r = 0; r < 8; ++r)
            __builtin_nontemporal_store(
                c[r], out + (size_t)(g0 + laneHi * 8 + r) * MSA_EMB + cb);
    }
}

// ---------------------------------------------------------------------------
// Pairwise: out[i,j,:] = q[j] + k[i] + posrow[clip(j-i,-32,32)+32]
// One block per i; 256 threads = 8 j-lanes x 32 float4 channels.
// ---------------------------------------------------------------------------
__global__ __launch_bounds__(256) void pair_kernel(
    const float* __restrict__ q, const float* __restrict__ k,
    const float* __restrict__ posrow, float* __restrict__ out)
{
    const int i   = blockIdx.x;
    const int tid = threadIdx.x;
    const int c4  = tid & 31;          // 128 channels / 4
    const int j0  = tid >> 5;

    const f4 kv = ((const f4*)(k + i * PAIR_DIM))[c4];
    float* orow = out + (size_t)i * LSEQ * PAIR_DIM;

    for (int j = j0; j < LSEQ; j += 8) {
        int rel = j - i;
        rel = rel < -32 ? -32 : (rel > 32 ? 32 : rel);
        rel += 32;
        const f4 qv = ((const f4*)(q + j * PAIR_DIM))[c4];
        const f4 pv = ((const f4*)(posrow + rel * PAIR_DIM))[c4];
        const f4 r = qv + kv + pv;
        __builtin_nontemporal_store(r, (f4*)(orow + (size_t)j * PAIR_DIM) + c4);
    }
}

// ---------------------------------------------------------------------------
extern "C" void kernel_launch(void* const* d_in, const int* in_sizes, int n_in,
                              void* d_out, int out_size, void* d_ws, size_t ws_size,
                              hipStream_t stream) {
    (void)in_sizes; (void)n_in; (void)out_size; (void)ws_size;

    const float* seq   = (const float*)d_in[0];
    const float* msa   = (const float*)d_in[1];
    const float* Wmsa  = (const float*)d_in[2];
    const float* bmsa  = (const float*)d_in[3];
    const float* Wq    = (const float*)d_in[4];
    const float* bq    = (const float*)d_in[5];
    const float* Wk    = (const float*)d_in[6];
    const float* bk    = (const float*)d_in[7];
    const float* Ws    = (const float*)d_in[8];
    const float* bs    = (const float*)d_in[9];
    const float* Wpos  = (const float*)d_in[10];
    const float* bpos  = (const float*)d_in[11];
    const float* Wpos2 = (const float*)d_in[12];
    const float* bpos2 = (const float*)d_in[13];

    float* ws       = (float*)d_ws;
    float* combined = ws;                               // 1024*256
    float* q        = combined + LSEQ * MSA_EMB;        // 1024*128
    float* k        = q + LSEQ * PAIR_DIM;              // 1024*128
    float* posrow   = k + LSEQ * PAIR_DIM;              // 65*128   (~2.03 MB total)

    float* msa_out  = (float*)d_out;                          // 256*1024*256
    float* pair_out = msa_out + (size_t)N_MSA * LSEQ * MSA_EMB;

    combined_kernel<<<LSEQ, 256, 0, stream>>>(seq, Ws, bs, Wpos2, bpos2, bmsa, combined);
    qk_kernel<<<LSEQ, 128, 0, stream>>>(seq, Wq, bq, Wk, bk, q, k);
    posrow_kernel<<<POS_BINS, 128, 0, stream>>>(Wpos, bpos, posrow);

    msa_wmma_kernel<<<(N_MSA * LSEQ) / (16 * 8), 256, 0, stream>>>(msa, Wmsa, combined, msa_out);
    pair_kernel<<<LSEQ, 256, 0, stream>>>(q, k, posrow, pair_out);
}